// SRSChannelEstimator_28252294873503
// MI455X (gfx1250) — compile-verified
//
#include <hip/hip_runtime.h>

#define S   8
#define BB  64
#define A   4
#define LL  3264
#define KK  12
#define RQ  816          // LL / LOCC
#define NCAND 21
#define MPAD 32
#define DELAY_MIN (-10)

typedef __attribute__((ext_vector_type(2))) float v2f;
typedef __attribute__((ext_vector_type(8))) float v8f;

#define C2PI_L ((float)(2.0 * 3.14159265358979323846 / (double)LL))

__device__ __forceinline__ int posmod(int x, int mdl) {
  int r = x % mdl;
  return r < 0 ? r + mdl : r;
}

// ---------------------------------------------------------------------------
// K0: build per-stream phasor matrix E[s][j][k] = exp(+2*pi*i * k * n_j / L),
// n_j = ideal_peak[s] + DELAY_MIN + j, rows j>=21 zero-padded (to 32 rows).
// ---------------------------------------------------------------------------
__global__ void k_phasor(const int* __restrict__ cyc,
                         float* __restrict__ Ere, float* __restrict__ Eim) {
  int idx = blockIdx.x * blockDim.x + threadIdx.x;   // s*MPAD*LL + j*LL + k
  if (idx >= S * MPAD * LL) return;
  int k = idx % LL;
  int j = (idx / LL) % MPAD;
  int s = idx / (LL * MPAD);
  float re = 0.0f, im = 0.0f;
  if (j < NCAND) {
    int ip = ((KK - cyc[s]) % KK) * (LL / KK);
    int n  = posmod(ip + DELAY_MIN + j, LL);
    int p  = (k * n) % LL;                  // exact, < 2^24
    __sincosf(C2PI_L * (float)p, &im, &re);
  }
  Ere[idx] = re;
  Eim[idx] = im;
}

// ---------------------------------------------------------------------------
// K1: candidate-power GEMM via V_WMMA_F32_16X16X4_F32.
// Per stream: H = E(32x3264) * X(3264x256) complex; cand_pow = sum_a |H|^2.
// One wave per 16x16 output tile; grid = (ntile=16, mtile=2, s=8).
// ---------------------------------------------------------------------------
__global__ void k_candpow(const float* __restrict__ lsr, const float* __restrict__ lsi,
                          const float* __restrict__ Ere, const float* __restrict__ Eim,
                          float* __restrict__ cand) {
  int ntile = blockIdx.x;        // 16 column tiles of 16  (cols = b*4+a)
  int mtile = blockIdx.y;        // 2 row tiles of 16      (rows = candidate j)
  int s     = blockIdx.z;        // stream
  int lane  = threadIdx.x;       // 0..31 (wave32)
  int lo = lane & 15, hi = lane >> 4;

  // A tile: lane holds row M = lo; VGPR p holds K = p + 2*hi  -> float2 of consecutive k
  const float* er = Ere + ((size_t)(s * MPAD + mtile * 16 + lo)) * LL + 2 * hi;
  const float* ei = Eim + ((size_t)(s * MPAD + mtile * 16 + lo)) * LL + 2 * hi;
  // B tile: lane holds col N = lo; VGPR p holds K = p + 2*hi  -> float2 of consecutive k
  int col = ntile * 16 + lo;     // (b*4 + a)
  const float* xr = lsr + ((size_t)(s * BB * A + col)) * LL + 2 * hi;
  const float* xi = lsi + ((size_t)(s * BB * A + col)) * LL + 2 * hi;

  v8f arr = {};  // Er*Xr
  v8f aii = {};  // Ei*Xi
  v8f ari = {};  // Er*Xi
  v8f air = {};  // Ei*Xr
  for (int k0 = 0; k0 < LL; k0 += 4) {
    v2f aRe = *(const v2f*)(er + k0);
    v2f aIm = *(const v2f*)(ei + k0);
    v2f bRe = *(const v2f*)(xr + k0);
    v2f bIm = *(const v2f*)(xi + k0);
    arr = __builtin_amdgcn_wmma_f32_16x16x4_f32(false, aRe, false, bRe, (short)0, arr, false, false);
    aii = __builtin_amdgcn_wmma_f32_16x16x4_f32(false, aIm, false, bIm, (short)0, aii, false, false);
    ari = __builtin_amdgcn_wmma_f32_16x16x4_f32(false, aRe, false, bIm, (short)0, ari, false, false);
    air = __builtin_amdgcn_wmma_f32_16x16x4_f32(false, aIm, false, bRe, (short)0, air, false, false);
  }

  // D tile: lane holds col N = lo, VGPR v holds row j = mtile*16 + 8*hi + v.
  // Reduce |H|^2 over antenna a = col&3 via xor-1/xor-2 shuffles (stay in 16-lane half).
  float pw[8];
#pragma unroll
  for (int v = 0; v < 8; ++v) {
    float hr = arr[v] - aii[v];
    float hm = ari[v] + air[v];
    float p  = hr * hr + hm * hm;
    p += __shfl_xor(p, 1, 32);
    p += __shfl_xor(p, 2, 32);
    pw[v] = p;
  }
  if ((lane & 3) == 0) {
    int b     = ntile * 4 + (lo >> 2);
    int jbase = mtile * 16 + 8 * hi;
#pragma unroll
    for (int v = 0; v < 8; ++v)
      cand[((size_t)(s * BB + b)) * MPAD + jbase + v] = pw[v];
  }
}

// ---------------------------------------------------------------------------
// K2: argmax over 21 candidates (first-max, matching jnp.argmax); t_off, m.
// ---------------------------------------------------------------------------
__global__ void k_argmax(const int* __restrict__ cyc, const float* __restrict__ cand,
                         int* __restrict__ m_ws, int* __restrict__ toff_ws) {
  int idx = blockIdx.x * blockDim.x + threadIdx.x;   // (s,b)
  if (idx >= S * BB) return;
  int s = idx / BB;
  const float* c = cand + (size_t)idx * MPAD;
  int best = 0; float bv = c[0];
  for (int j = 1; j < NCAND; ++j) { float v = c[j]; if (v > bv) { bv = v; best = j; } }
  int toff = DELAY_MIN + best;
  int ip   = ((KK - cyc[s]) % KK) * (LL / KK);
  toff_ws[idx] = toff;
  m_ws[idx]    = toff + ip;
}

// ---------------------------------------------------------------------------
// K3: h_avg[s,b,a,r] = mean_{c<4} ls[..,4r+c] * exp(+2*pi*i*m*(4r+c)/L)
// ---------------------------------------------------------------------------
__global__ void k_havg(const float* __restrict__ lsr, const float* __restrict__ lsi,
                       const int* __restrict__ m_ws, float2* __restrict__ havg) {
  int idx = blockIdx.x * blockDim.x + threadIdx.x;   // (s,b,a,r)
  if (idx >= S * BB * A * RQ) return;
  int r   = idx % RQ;
  int sba = idx / RQ;
  int s   = sba / (BB * A);
  int b   = (sba / A) % BB;
  int mn  = posmod(m_ws[s * BB + b], LL);
  const float4 lr = *(const float4*)(lsr + (size_t)sba * LL + 4 * r);
  const float4 li = *(const float4*)(lsi + (size_t)sba * LL + 4 * r);
  float lre[4] = {lr.x, lr.y, lr.z, lr.w};
  float lim[4] = {li.x, li.y, li.z, li.w};
  float accr = 0.0f, acci = 0.0f;
#pragma unroll
  for (int c = 0; c < 4; ++c) {
    int k = 4 * r + c;
    int p = (k * mn) % LL;
    float sn, cs; __sincosf(C2PI_L * (float)p, &sn, &cs);
    accr += lre[c] * cs - lim[c] * sn;
    acci += lre[c] * sn + lim[c] * cs;
  }
  havg[idx] = make_float2(accr * 0.25f, acci * 0.25f);
}

// ---------------------------------------------------------------------------
// K4: residual[b,a,l] = ls[0,b,a,l] - sum_s h_interp(s)*conj(ph_m(s))
// ---------------------------------------------------------------------------
__global__ void k_residual(const float* __restrict__ lsr, const float* __restrict__ lsi,
                           const float2* __restrict__ havg, const int* __restrict__ m_ws,
                           float2* __restrict__ resid) {
  int idx = blockIdx.x * blockDim.x + threadIdx.x;   // (b,a,l)
  if (idx >= BB * A * LL) return;
  int l  = idx % LL;
  int ba = idx / LL;
  int b  = ba / A;
  float t = ((float)l - 1.5f) * 0.25f;
  t = fminf(fmaxf(t, 0.0f), (float)(RQ - 1));
  int i0 = (int)t;
  int i1 = min(i0 + 1, RQ - 1);
  float fr = t - (float)i0;
  float accr = 0.0f, acci = 0.0f;
#pragma unroll
  for (int s = 0; s < S; ++s) {
    size_t hb = ((size_t)s * BB * A + ba) * RQ;
    float2 h0 = havg[hb + i0], h1 = havg[hb + i1];
    float hr = h0.x + (h1.x - h0.x) * fr;
    float hi = h0.y + (h1.y - h0.y) * fr;
    int mn = posmod(m_ws[s * BB + b], LL);
    int p  = (l * mn) % LL;
    float sn, cs; __sincosf(C2PI_L * (float)p, &sn, &cs);
    accr += hr * cs + hi * sn;          // h * conj(ph_m)
    acci += hi * cs - hr * sn;
  }
  resid[idx] = make_float2(lsr[idx] - accr, lsi[idx] - acci);   // ls[0] slice
}

// ---------------------------------------------------------------------------
// K5: final = (h_interp + residual*ph_m) * conj(ph_T)  -> out[2,S,B,A,L]
// ---------------------------------------------------------------------------
__global__ void k_final(const float2* __restrict__ havg, const float2* __restrict__ resid,
                        const int* __restrict__ m_ws, const int* __restrict__ toff_ws,
                        float* __restrict__ out) {
  int idx = blockIdx.x * blockDim.x + threadIdx.x;   // (s,b,a,l)
  if (idx >= S * BB * A * LL) return;
  int l   = idx % LL;
  int sba = idx / LL;
  int s   = sba / (BB * A);
  int b   = (sba / A) % BB;
  int ba  = sba % (BB * A);
  float t = ((float)l - 1.5f) * 0.25f;
  t = fminf(fmaxf(t, 0.0f), (float)(RQ - 1));
  int i0 = (int)t;
  int i1 = min(i0 + 1, RQ - 1);
  float fr = t - (float)i0;
  size_t hb = (size_t)sba * RQ;
  float2 h0 = havg[hb + i0], h1 = havg[hb + i1];
  float hr = h0.x + (h1.x - h0.x) * fr;
  float hi = h0.y + (h1.y - h0.y) * fr;
  float2 rs = resid[(size_t)ba * LL + l];
  int sb = s * BB + b;
  int mn = posmod(m_ws[sb], LL);
  int tn = posmod(toff_ws[sb], LL);
  float smv, cmv, stv, ctv;
  __sincosf(C2PI_L * (float)((l * mn) % LL), &smv, &cmv);
  __sincosf(C2PI_L * (float)((l * tn) % LL), &stv, &ctv);
  float wr = hr + rs.x * cmv - rs.y * smv;   // h + res*ph_m
  float wi = hi + rs.x * smv + rs.y * cmv;
  float fre = wr * ctv + wi * stv;           // * conj(ph_T)
  float fim = wi * ctv - wr * stv;
  out[idx] = fre;
  out[(size_t)S * BB * A * LL + idx] = fim;
}

// ---------------------------------------------------------------------------
extern "C" void kernel_launch(void* const* d_in, const int* in_sizes, int n_in,
                              void* d_out, int out_size, void* d_ws, size_t ws_size,
                              hipStream_t stream) {
  (void)in_sizes; (void)n_in; (void)out_size; (void)ws_size;
  const float* lsr = (const float*)d_in[0];
  const float* lsi = (const float*)d_in[1];
  const int*   cyc = (const int*)d_in[2];
  float* out = (float*)d_out;

  char* ws = (char*)d_ws;
  size_t o = 0;
  float*  Ere    = (float*)(ws + o);  o += (size_t)S * MPAD * LL * sizeof(float);
  float*  Eim    = (float*)(ws + o);  o += (size_t)S * MPAD * LL * sizeof(float);
  float*  cand   = (float*)(ws + o);  o += (size_t)S * BB * MPAD * sizeof(float);
  int*    m_ws   = (int*)(ws + o);    o += (size_t)S * BB * sizeof(int);
  int*    toffws = (int*)(ws + o);    o += (size_t)S * BB * sizeof(int);
  float2* havg   = (float2*)(ws + o); o += (size_t)S * BB * A * RQ * sizeof(float2);
  float2* resid  = (float2*)(ws + o); o += (size_t)BB * A * LL * sizeof(float2);

  { int n = S * MPAD * LL;
    k_phasor<<<(n + 255) / 256, 256, 0, stream>>>(cyc, Ere, Eim); }

  k_candpow<<<dim3(16, 2, 8), 32, 0, stream>>>(lsr, lsi, Ere, Eim, cand);

  k_argmax<<<(S * BB + 255) / 256, 256, 0, stream>>>(cyc, cand, m_ws, toffws);

  { int n = S * BB * A * RQ;
    k_havg<<<(n + 255) / 256, 256, 0, stream>>>(lsr, lsi, m_ws, havg); }

  { int n = BB * A * LL;
    k_residual<<<(n + 255) / 256, 256, 0, stream>>>(lsr, lsi, havg, m_ws, resid); }

  { int n = S * BB * A * LL;
    k_final<<<(n + 255) / 256, 256, 0, stream>>>(havg, resid, m_ws, toffws, out); }
}